// MSECannyLoss_56418690400988
// MI455X (gfx1250) — compile-verified
//
#include <hip/hip_runtime.h>
#include <math.h>
#include <stdint.h>

typedef float v2f __attribute__((ext_vector_type(2)));
typedef float v8f __attribute__((ext_vector_type(8)));

#define IMG_H 512
#define IMG_W 512
#define NB    16
#define NCH   3

#if defined(__has_builtin)
#if __has_builtin(__builtin_amdgcn_global_load_async_to_lds_b128)
#define HAVE_ASYNC_LDS 1
#endif
#endif

#ifdef HAVE_ASYNC_LDS
// match the builtin's parameter type exactly: int __attribute__((vector_size(16)))
typedef int v4i_raw __attribute__((vector_size(4 * sizeof(int))));
typedef __attribute__((address_space(1))) v4i_raw* gv4p;   // global v4i*
typedef __attribute__((address_space(3))) float*   lfp;    // LDS float*
typedef __attribute__((address_space(3))) v4i_raw* lv4p;   // LDS v4i*
#endif

// 5-tap gaussian (sigma=1), normalized: exp(-t^2/2)/sum, t=-2..2
__device__ __forceinline__ float gw(int k) { // k in [0,4]
    return (k == 0 || k == 4) ? 0.054488684549643f
         : (k == 1 || k == 3) ? 0.244201342003233f
         : 0.402619946894248f;
}

__device__ __forceinline__ int reflect_idx(int i) {
    if (i < 0) i = -i;                 // jnp 'reflect': -1 -> 1
    if (i > IMG_H - 1) i = 2 * (IMG_H - 1) - i; // 512 -> 510
    return i;
}

// ---------------- kernel A: gray min/max partials -----------------
__global__ __launch_bounds__(256)
void k_gray_minmax(const float* __restrict__ tgt, float* __restrict__ mn_part,
                   float* __restrict__ mx_part) {
    __shared__ float smn[256], smx[256];
    const long long n = (long long)NB * IMG_H * IMG_W;
    const long long stride = (long long)gridDim.x * blockDim.x;
    float mn = 3.402823466e38f, mx = -3.402823466e38f;
    for (long long i = (long long)blockIdx.x * blockDim.x + threadIdx.x; i < n; i += stride) {
        long long b   = i / (IMG_H * IMG_W);
        long long pix = i % (IMG_H * IMG_W);
        const float* tb = tgt + b * (long long)NCH * IMG_H * IMG_W + pix;
        float g = 0.299f * tb[0] + 0.587f * tb[IMG_H * IMG_W] + 0.114f * tb[2 * IMG_H * IMG_W];
        mn = fminf(mn, g);
        mx = fmaxf(mx, g);
    }
    smn[threadIdx.x] = mn; smx[threadIdx.x] = mx;
    __syncthreads();
    for (int o = 128; o > 0; o >>= 1) {
        if (threadIdx.x < o) {
            smn[threadIdx.x] = fminf(smn[threadIdx.x], smn[threadIdx.x + o]);
            smx[threadIdx.x] = fmaxf(smx[threadIdx.x], smx[threadIdx.x + o]);
        }
        __syncthreads();
    }
    if (threadIdx.x == 0) { mn_part[blockIdx.x] = smn[0]; mx_part[blockIdx.x] = smx[0]; }
}

// ---------------- kernel B: finalize min/max ----------------------
__global__ __launch_bounds__(256)
void k_minmax_final(const float* __restrict__ mn_part, const float* __restrict__ mx_part,
                    float* __restrict__ ws01, int npart) {
    __shared__ float smn[256], smx[256];
    float mn = 3.402823466e38f, mx = -3.402823466e38f;
    for (int i = threadIdx.x; i < npart; i += 256) {
        mn = fminf(mn, mn_part[i]);
        mx = fmaxf(mx, mx_part[i]);
    }
    smn[threadIdx.x] = mn; smx[threadIdx.x] = mx;
    __syncthreads();
    for (int o = 128; o > 0; o >>= 1) {
        if (threadIdx.x < o) {
            smn[threadIdx.x] = fminf(smn[threadIdx.x], smn[threadIdx.x + o]);
            smx[threadIdx.x] = fmaxf(smx[threadIdx.x], smx[threadIdx.x + o]);
        }
        __syncthreads();
    }
    if (threadIdx.x == 0) { ws01[0] = smn[0]; ws01[1] = smx[0]; }
}

// ---------------- kernel C: fused WMMA blur + sobel + NMS + loss ---
// One wave32 per 16x16 output tile. Gaussian blur done on the matrix pipe
// as banded-matrix products with V_WMMA_F32_16X16X4_F32. The interior
// output/target pixels are prefetched to LDS via async-to-LDS (ASYNCcnt)
// at kernel entry and consumed only after all stencil work completes.
__global__ __launch_bounds__(32)
void k_edge_loss(const float* __restrict__ outp, const float* __restrict__ tgt,
                 const float* __restrict__ ws01, float* __restrict__ loss_part) {
    __shared__ float g[24][25];   // normalized gray, halo 4 (reflect)
    __shared__ float V[20][25];   // vertical-blurred, rows y0-2..y0+17, cols x0-4..x0+19
    __shared__ float Bl[20][21];  // fully blurred, halo 2 region
    __shared__ float m[18][19];   // magnitude, halo 1
#ifdef HAVE_ASYNC_LDS
    __shared__ __align__(16) float so[3][256];  // interior output tile
    __shared__ __align__(16) float st[3][256];  // interior target tile
#endif

    const int t  = threadIdx.x;      // 0..31
    const int lm = t & 15;           // lane within half (row/col index)
    const int lh = t >> 4;           // half select
    const int x0 = blockIdx.x * 16, y0 = blockIdx.y * 16, bi = blockIdx.z;

    const float* tb = tgt  + (size_t)bi * NCH * IMG_H * IMG_W;
    const float* ob = outp + (size_t)bi * NCH * IMG_H * IMG_W;

#ifdef HAVE_ASYNC_LDS
    // async prefetch: each lane copies 2 x 16B per channel per tensor
    {
#pragma unroll
        for (int j = 0; j < 2; ++j) {
            int li = 4 * (t + 32 * j);            // 0..252 step 4 (16B aligned)
            int r = li >> 4, c = li & 15;
            size_t p = (size_t)(y0 + r) * IMG_W + (x0 + c);
#pragma unroll
            for (int ch = 0; ch < NCH; ++ch) {
                __builtin_amdgcn_global_load_async_to_lds_b128(
                    (gv4p)(uintptr_t)(ob + p + (size_t)ch * IMG_H * IMG_W),
                    (lv4p)(lfp)&so[ch][li], 0, 0);
                __builtin_amdgcn_global_load_async_to_lds_b128(
                    (gv4p)(uintptr_t)(tb + p + (size_t)ch * IMG_H * IMG_W),
                    (lv4p)(lfp)&st[ch][li], 0, 0);
            }
        }
    }
#endif

    const float gmin = ws01[0];
    const float ginv = 1.0f / (ws01[1] - gmin + 1e-8f);

    // ---- load gray halo tile (reflect pad), normalize ----
    for (int i = 0; i < 18; ++i) {
        int idx = t + 32 * i;
        if (idx < 24 * 24) {
            int r = idx / 24, c = idx % 24;
            int ya = reflect_idx(y0 - 4 + r);
            int xa = reflect_idx(x0 - 4 + c);
            size_t p = (size_t)ya * IMG_W + xa;
            float gr = 0.299f * tb[p] + 0.587f * tb[p + IMG_H * IMG_W]
                     + 0.114f * tb[p + 2 * IMG_H * IMG_W];
            g[r][c] = (gr - gmin) * ginv;
        }
    }
    __syncthreads();

    // ---- vertical blur: V(20x24) = S(20x24, banded) x g(24x24) ----
    // A frag (16x4 f32): lane l -> M = l%16 ; K = v + 2*(l/16)
    // B frag (4x16 f32): lane l -> N = l%16 ; K = v + 2*(l/16)
    // D frag (16x16 f32): lane l, VGPR v -> M = v + 8*(l/16), N = l%16
    for (int R = 0; R < 32; R += 16) {
        for (int n0 = 0; n0 < 32; n0 += 16) {
            v8f acc = {0.f, 0.f, 0.f, 0.f, 0.f, 0.f, 0.f, 0.f};
            for (int kc = 0; kc < 6; ++kc) {
                v2f a, b;
#pragma unroll
                for (int v = 0; v < 2; ++v) {
                    int j  = 4 * kc + v + 2 * lh;   // K index (gray row)
                    int is = R + lm;                 // S row
                    int kd = j - is;                 // band offset
                    a[v] = (is < 20 && kd >= 0 && kd <= 4) ? gw(kd) : 0.0f;
                    int col = n0 + lm;
                    b[v] = (col < 24) ? g[j][col] : 0.0f;
                }
                acc = __builtin_amdgcn_wmma_f32_16x16x4_f32(
                    false, a, false, b, (short)0, acc, false, false);
            }
#pragma unroll
            for (int v = 0; v < 8; ++v) {
                int row = R + v + 8 * lh, col = n0 + lm;
                if (row < 20 && col < 24) V[row][col] = acc[v];
            }
        }
    }
    __syncthreads();

    // ---- horizontal blur: Bl(20x20) = V(20x24) x T(24x20, banded) ----
    for (int R = 0; R < 32; R += 16) {
        for (int n0 = 0; n0 < 32; n0 += 16) {
            v8f acc = {0.f, 0.f, 0.f, 0.f, 0.f, 0.f, 0.f, 0.f};
            for (int kc = 0; kc < 6; ++kc) {
                v2f a, b;
#pragma unroll
                for (int v = 0; v < 2; ++v) {
                    int j   = 4 * kc + v + 2 * lh;  // K index (V col)
                    int row = R + lm;
                    a[v] = (row < 20 && j < 24) ? V[row][j] : 0.0f;
                    int c  = n0 + lm;
                    int kd = j - c;
                    b[v] = (kd >= 0 && kd <= 4) ? gw(kd) : 0.0f;
                }
                acc = __builtin_amdgcn_wmma_f32_16x16x4_f32(
                    false, a, false, b, (short)0, acc, false, false);
            }
#pragma unroll
            for (int v = 0; v < 8; ++v) {
                int row = R + v + 8 * lh, col = n0 + lm;
                if (row < 20 && col < 20) Bl[row][col] = acc[v];
            }
        }
    }
    __syncthreads();

    // ---- sobel + magnitude (halo 1, replicate-pad blurred) ----
    for (int i = 0; i < 11; ++i) {
        int idx = t + 32 * i;
        if (idx < 18 * 18) {
            int p = idx / 18, q = idx % 18;
            int ya = y0 - 1 + p, xa = x0 - 1 + q;
            auto blv = [&](int dy, int dx) {
                int rr = min(max(ya + dy, 0), IMG_H - 1) - (y0 - 2);
                int cc = min(max(xa + dx, 0), IMG_W - 1) - (x0 - 2);
                return Bl[rr][cc];
            };
            float gx = (blv(-1, 1) - blv(-1, -1)) + 2.f * (blv(0, 1) - blv(0, -1))
                     + (blv(1, 1) - blv(1, -1));
            float gy = (blv(1, -1) + 2.f * blv(1, 0) + blv(1, 1))
                     - (blv(-1, -1) + 2.f * blv(-1, 0) + blv(-1, 1));
            m[p][q] = sqrtf(gx * gx + gy * gy + 1e-6f);
        }
    }
    __syncthreads();

#ifdef HAVE_ASYNC_LDS
    // prefetched interior tiles must be in LDS before the loss loop
#if __has_builtin(__builtin_amdgcn_s_wait_asynccnt)
    __builtin_amdgcn_s_wait_asynccnt(0);
#else
    asm volatile("s_wait_asynccnt 0" ::: "memory");
#endif
    asm volatile("" ::: "memory");
#endif

    // ---- NMS + weighted MSE over interior 16x16 ----
    const int DY[8] = {0, 1, 1, 1, 0, -1, -1, -1};
    const int DX[8] = {1, 1, 0, -1, -1, -1, 0, 1};
    float lsum = 0.f;
    for (int i = 0; i < 8; ++i) {
        int idx = t + 32 * i;          // 0..255
        int r = idx >> 4, c = idx & 15;
        int y = y0 + r, x = x0 + c;
        float mag = m[r + 1][c + 1];
        auto blc = [&](int dy, int dx) {
            int rr = min(max(y + dy, 0), IMG_H - 1) - (y0 - 2);
            int cc = min(max(x + dx, 0), IMG_W - 1) - (x0 - 2);
            return Bl[rr][cc];
        };
        float gx = (blc(-1, 1) - blc(-1, -1)) + 2.f * (blc(0, 1) - blc(0, -1))
                 + (blc(1, 1) - blc(1, -1));
        float gy = (blc(1, -1) + 2.f * blc(1, 0) + blc(1, 1))
                 - (blc(-1, -1) + 2.f * blc(-1, 0) + blc(-1, 1));
        // ang45 = round(rad2deg(atan2)/45) = round(atan2 * 4/pi), k in [-4,4]
        int k = (int)rintf(atan2f(gy, gx) * 1.27323954473516f);
        int pidx = (k + 8) & 7;
        int nidx = (k + 12) & 7;
        auto nmsdiff = [&](int d) {
            int yy = y + DY[d], xx = x + DX[d];
            float mv = (yy >= 0 && yy < IMG_H && xx >= 0 && xx < IMG_W)
                         ? m[r + 1 + DY[d]][c + 1 + DX[d]] : 0.f;
            return mag - mv;   // zero-padded neighbor
        };
        float selp = nmsdiff(pidx), seln = nmsdiff(nidx);
        float edge = (fminf(selp, seln) > 0.f) ? mag : 0.f;
        float wgt = 1.f + 2.0f * edge;   // EDGE_WEIGHT = 2.0
#ifdef HAVE_ASYNC_LDS
#pragma unroll
        for (int ch = 0; ch < NCH; ++ch) {
            float d = so[ch][idx] - st[ch][idx];
            lsum += d * d * wgt;
        }
#else
        size_t p = (size_t)y * IMG_W + x;
#pragma unroll
        for (int ch = 0; ch < NCH; ++ch) {
            float d = ob[p + (size_t)ch * IMG_H * IMG_W] - tb[p + (size_t)ch * IMG_H * IMG_W];
            lsum += d * d * wgt;
        }
#endif
    }
    // wave32 reduction (deterministic)
    for (int off = 16; off > 0; off >>= 1) lsum += __shfl_xor(lsum, off);
    if (t == 0)
        loss_part[((size_t)blockIdx.z * gridDim.y + blockIdx.y) * gridDim.x + blockIdx.x] = lsum;
}

// ---------------- kernel D: fixed-order final sum -> mean ---------
__global__ __launch_bounds__(256)
void k_final(const float* __restrict__ part, int n, float* __restrict__ out) {
    __shared__ float s[256];
    float a = 0.f;
    for (int i = threadIdx.x; i < n; i += 256) a += part[i];   // fixed stride order
    s[threadIdx.x] = a;
    __syncthreads();
    for (int o = 128; o > 0; o >>= 1) {
        if (threadIdx.x < o) s[threadIdx.x] += s[threadIdx.x + o];
        __syncthreads();
    }
    if (threadIdx.x == 0)
        out[0] = s[0] * (1.0f / ((float)NB * NCH * IMG_H * IMG_W));
}

extern "C" void kernel_launch(void* const* d_in, const int* in_sizes, int n_in,
                              void* d_out, int out_size, void* d_ws, size_t ws_size,
                              hipStream_t stream) {
    const float* outp = (const float*)d_in[0];
    const float* tgtp = (const float*)d_in[1];
    float* ws = (float*)d_ws;
    // ws layout (floats): [0]=gmin [1]=gmax [2..1025]=min parts
    // [1026..2049]=max parts  [4096..20479]=per-tile loss partials
    float* mn_part   = ws + 2;
    float* mx_part   = ws + 2 + 1024;
    float* loss_part = ws + 4096;

    k_gray_minmax<<<1024, 256, 0, stream>>>(tgtp, mn_part, mx_part);
    k_minmax_final<<<1, 256, 0, stream>>>(mn_part, mx_part, ws, 1024);
    dim3 grid(IMG_W / 16, IMG_H / 16, NB);   // 32 x 32 x 16 tiles
    k_edge_loss<<<grid, 32, 0, stream>>>(outp, tgtp, ws, loss_part);
    k_final<<<1, 256, 0, stream>>>(loss_part, 32 * 32 * NB, (float*)d_out);
}